// CustomComputationLayer_87368224735397
// MI455X (gfx1250) — compile-verified
//
#include <hip/hip_runtime.h>
#include <cstdint>

// ---------------------------------------------------------------------------
// Types for the gfx1250 TDM builtin (amdgpu-toolchain / clang-23: 6-arg form)
//   (uint32x4 g0, int32x8 g1, int32x4 g2, int32x4 g3, int32x8 g4, i32 cpol)
// ---------------------------------------------------------------------------
typedef __attribute__((ext_vector_type(4))) unsigned int v4u;
typedef __attribute__((ext_vector_type(8))) int          v8i;
typedef __attribute__((ext_vector_type(4))) int          v4i;

constexpr int T_     = 20;
constexpr int ROWLEN = 100;   // 5*T floats per row
constexpr int BLK    = 128;   // 4 wave32 waves; 51,200 B LDS tile -> 6 blocks/WGP

// ---------------------------------------------------------------------------
// Compile-time discount tables (match the numpy/jax constant setup).
//   D_t      = 0.995^t
//   pd_t     = linspace(0.001, 0.01, 21)  (step 0.00045)
//   Pbard_t  = 1 - cumsum(pd_t)
//   PBAR_D   = Pbard_t * D_t ; P_D = (pd_t * D_t)[:-1]
// We fold the factor 2 (from v(x)=x^0.5/0.5) and UTILITY_FACTOR=1/T^2 in.
// ---------------------------------------------------------------------------
struct Tab {
  float cw[T_];     // 2 * PBAR_D[t] / T^2      (consumption weight)
  float lw[T_];     // 2 * P_D[t]    / T^2      (legacy weight)
  float invpd[T_];  // 1 / (ALPHA_SCALE*pd_t[t] + 1e-6)
  float tail;       // 2 * PBAR_D[T] / T^2
};

constexpr Tab make_tab() {
  Tab tb{};
  double D = 1.0, cum = 0.0, tail = 0.0;
  const double UF = 1.0 / (double)(T_ * T_);
  for (int t = 0; t <= T_; ++t) {
    const double pd = 0.001 + t * (0.009 / 20.0);
    cum += pd;
    const double pbar = (1.0 - cum) * D;
    if (t < T_) {
      tb.cw[t]    = (float)(2.0 * pbar * UF);
      tb.lw[t]    = (float)(2.0 * pd * D * UF);
      tb.invpd[t] = (float)(1.0 / (pd + 1e-6));
    } else {
      tail = 2.0 * pbar * UF;
    }
    D *= 0.995;
  }
  tb.tail = (float)tail;
  return tb;
}

__device__ constexpr Tab TAB = make_tab();

__device__ __forceinline__ float clampP(float v)  { return fminf(fmaxf(v, 1e-6f), 1e6f); }
__device__ __forceinline__ float clamp01(float v) { return fminf(fmaxf(v, 1e-6f), 1.0f - 1e-6f); }
__device__ __forceinline__ float fexp(float x)    { return __builtin_amdgcn_exp2f(x * 1.44269504088896f); }

// ---------------------------------------------------------------------------
// One thread = one row. Tile of BLK rows staged into LDS by the Tensor Data
// Mover (one descriptor per block, issued by wave 0), then each thread walks
// its contiguous 100-float row out of LDS with ds_load_b128.
// ---------------------------------------------------------------------------
__global__ void __launch_bounds__(BLK)
lifecycle_utility_kernel(const float* __restrict__ in, float* __restrict__ out, int B) {
  __shared__ float smem[BLK * ROWLEN];   // 51,200 bytes

  const int tid = threadIdx.x;
  const long long row0 = (long long)blockIdx.x * BLK;

  if (tid < 32) {  // wave 0 issues the TDM load for the whole tile
    const unsigned long long ga =
        (unsigned long long)(uintptr_t)in + (unsigned long long)row0 * (ROWLEN * 4ull);
    const unsigned int lds = (unsigned int)(uintptr_t)(&smem[0]);

    long long rem = (long long)B - row0;
    int rows = rem >= BLK ? BLK : (rem < 0 ? 0 : (int)rem);

    // D# group 0: count=1 | lds_addr | global_addr[56:0] | type=2
    v4u g0;
    g0.x = 1u;
    g0.y = lds;
    g0.z = (unsigned int)ga;
    g0.w = ((unsigned int)(ga >> 32) & 0x01FFFFFFu) | 0x80000000u;

    // D# group 1: data_size=4B, tensor_dim0=100, tensor_dim1=rows,
    //             tile_dim0=100, tile_dim1=rows, tensor_dim0_stride=100
    v8i g1;
    g1.s0 = 0x00020000;                  // [17:16] data_size = 2 (4 bytes)
    g1.s1 = (int)(ROWLEN << 16);         // tensor_dim0 lo16 (barrier addr = 0)
    g1.s2 = (int)(rows   << 16);         // tensor_dim0 hi16 = 0 | tensor_dim1 lo16
    g1.s3 = (int)(ROWLEN << 16);         // tensor_dim1 hi16 = 0 | tile_dim0
    g1.s4 = rows;                        // tile_dim1 | tile_dim2 = 0
    g1.s5 = ROWLEN;                      // tensor_dim0_stride lo32
    g1.s6 = 0;                           // dim0_stride hi16 | dim1_stride lo16
    g1.s7 = 0;                           // dim1_stride hi32

    v4i g2 = {0, 0, 0, 0};               // 2-D tensor: groups 2/3 unused
    v4i g3 = {0, 0, 0, 0};
    v8i g4 = {0, 0, 0, 0, 0, 0, 0, 0};   // extra descriptor group (clang-23 form)

    __builtin_amdgcn_tensor_load_to_lds(g0, g1, g2, g3, g4, 0);
    __builtin_amdgcn_s_wait_tensorcnt(0);
  }
  __syncthreads();

  const float* __restrict__ row = &smem[tid * ROWLEN];

  float x = 1.0f;     // x_0
  float acc = 0.0f;   // sum of discounted utilities (weights pre-scaled)

#pragma unroll
  for (int t = 0; t < T_; ++t) {
    const float cr = row[4 * t + 0];
    const float br = row[4 * t + 1];
    const float ir = row[4 * t + 2];
    const float rr = row[4 * t + 3];
    const float il = row[4 * T_ + t];

    // softmax over (c,b,i) with max subtraction (matches jax.nn.softmax)
    const float m  = fmaxf(cr, fmaxf(br, ir));
    const float ec = fexp(cr - m);
    const float eb = fexp(br - m);
    const float ei = fexp(ir - m);
    const float inv = __builtin_amdgcn_rcpf(ec + eb + ei);
    const float c = clamp01(ec * inv);
    const float b = clamp01(eb * inv);

    // sigmoid(rho_raw)
    const float rho = clamp01(__builtin_amdgcn_rcpf(1.0f + fexp(-rr)));

    // wealth fraction for the recursion
    const float wf = clampP(1.02f - c - b + rho * il);

    // consumption utility: 2*sqrt(clip(c*x)) * PBAR_D[t] * UF
    acc += __builtin_amdgcn_sqrtf(clampP(c * x)) * TAB.cw[t];

    // legacy utility: 2*sqrt(clip(x*(1 + b/pd))) * P_D[t] * UF
    const float lt = x * fmaf(b, TAB.invpd[t], 1.0f);
    acc += __builtin_amdgcn_sqrtf(clampP(lt)) * TAB.lw[t];

    // x_{t+1} = x_t * wf + 1
    x = fmaf(x, wf, 1.0f);
  }

  // terminal bequest: 2*sqrt(clip(x_T)) * PBAR_D[T] * UF
  acc += __builtin_amdgcn_sqrtf(clampP(x)) * TAB.tail;

  const long long r = row0 + tid;
  if (r < (long long)B) out[r] = acc;
}

// ---------------------------------------------------------------------------
// Harness entry point
// ---------------------------------------------------------------------------
extern "C" void kernel_launch(void* const* d_in, const int* in_sizes, int n_in,
                              void* d_out, int out_size, void* d_ws, size_t ws_size,
                              hipStream_t stream) {
  (void)n_in; (void)d_ws; (void)ws_size; (void)out_size;
  const float* in = (const float*)d_in[0];
  float* out = (float*)d_out;
  const int B = in_sizes[0] / (5 * T_);
  const int blocks = (B + BLK - 1) / BLK;
  hipLaunchKernelGGL(lifecycle_utility_kernel, dim3(blocks), dim3(BLK), 0, stream,
                     in, out, B);
}